// SCAM_5274219839998
// MI455X (gfx1250) — compile-verified
//
#include <hip/hip_runtime.h>
#include <hip/hip_bf16.h>
#include <cmath>

#define Cc   64
#define Hh   48
#define Ww   256
#define HWp  (Hh * Ww)          // 12288 pixels
#define CHW  (Cc * HWp)         // 786432
#define DWIN 64
#define SCALE 0.125f            // 64^-0.5

typedef __attribute__((ext_vector_type(2))) float v2f;
typedef __attribute__((ext_vector_type(8))) float v8f;

__device__ __forceinline__ v8f wmma4(v2f a, v2f b, v8f c) {
    // D = A(16x4 f32) * B(4x16 f32) + C(16x16 f32)
    return __builtin_amdgcn_wmma_f32_16x16x4_f32(
        /*neg_a=*/false, a, /*neg_b=*/false, b,
        /*c_mod=*/(short)0, c, /*reuse_a=*/false, /*reuse_b=*/false);
}

// ---------------------------------------------------------------------------
// Kernel A: per-pixel LayerNorm over channels (biased variance, eps=1e-6)
// ---------------------------------------------------------------------------
__global__ __launch_bounds__(256)
void scam_ln_kernel(const float* __restrict__ xl, const float* __restrict__ xr,
                    const float* __restrict__ gl, const float* __restrict__ bl,
                    const float* __restrict__ gr, const float* __restrict__ br,
                    float* __restrict__ xn) {
    int p    = blockIdx.x * blockDim.x + threadIdx.x;   // pixel
    int side = blockIdx.y;
    if (p >= HWp) return;
    const float* x = side ? xr : xl;
    const float* g = side ? gr : gl;
    const float* b = side ? br : bl;
    float* out = xn + (size_t)side * CHW;

    float s = 0.f, ss = 0.f;
    #pragma unroll 8
    for (int c = 0; c < Cc; ++c) {
        float v = x[(size_t)c * HWp + p];
        s += v; ss += v * v;
    }
    float mean = s * (1.0f / Cc);
    float var  = ss * (1.0f / Cc) - mean * mean;
    float inv  = rsqrtf(var + 1e-6f);
    #pragma unroll 8
    for (int c = 0; c < Cc; ++c) {
        float v = (x[(size_t)c * HWp + p] - mean) * inv * g[c] + b[c];
        out[(size_t)c * HWp + p] = v;
    }
}

// ---------------------------------------------------------------------------
// Kernel B: 1x1 conv projections as WMMA f32 GEMM.
//   out[p, o] = sum_c X[c, p] * W[o, c] + bias[o]
// grid.x = HWp/16 pixel tiles, grid.y = 6 (q_l,k_l,v_l,q_r,k_r,v_r)
// block = 128 (4 waves); wave w -> out-channel tile o0 = 16*w
// output layout: (pixel-major, channel-contiguous) per buffer.
// ---------------------------------------------------------------------------
__global__ __launch_bounds__(128)
void scam_proj_kernel(const float* __restrict__ xl, const float* __restrict__ xr,
                      const float* __restrict__ xn,
                      const float* __restrict__ Wq, const float* __restrict__ bq,
                      const float* __restrict__ Wk, const float* __restrict__ bk,
                      const float* __restrict__ Wv, const float* __restrict__ bv,
                      float* __restrict__ qkv) {
    int lane = threadIdx.x & 31;
    int wv   = threadIdx.x >> 5;          // 0..3
    int half = lane >> 4;                 // 0/1
    int lm   = lane & 15;

    int ptile = blockIdx.x;               // 0..767
    int py    = blockIdx.y;               // 0..5
    int side  = py / 3;
    int proj  = py % 3;                   // 0=q 1=k 2=v

    const float* X  = (proj == 2) ? (side ? xr : xl) : (xn + (size_t)side * CHW);
    const float* Wm = (proj == 0) ? Wq : (proj == 1) ? Wk : Wv;
    const float* bs = (proj == 0) ? bq : (proj == 1) ? bk : bv;
    float* dst = qkv + (size_t)py * CHW;

    int p0 = ptile * 16;
    int o0 = wv * 16;

    v8f acc = {};
    #pragma unroll
    for (int s = 0; s < 16; ++s) {
        int c0 = 4 * s + 2 * half;
        v2f a, b;
        // A(m=lm, k) = X[c0+k, p0+lm]
        a.x = X[(size_t)(c0 + 0) * HWp + p0 + lm];
        a.y = X[(size_t)(c0 + 1) * HWp + p0 + lm];
        // B(k, n=lm) = Wm[o0+lm, c0+k]
        b.x = Wm[(o0 + lm) * Cc + c0 + 0];
        b.y = Wm[(o0 + lm) * Cc + c0 + 1];
        acc = wmma4(a, b, acc);
    }
    float bias = bs[o0 + lm];
    #pragma unroll
    for (int v = 0; v < 8; ++v) {
        int m = v + 8 * half;                              // pixel offset in tile
        dst[(size_t)(p0 + m) * Cc + o0 + lm] = acc[v] + bias;
    }
}

// ---------------------------------------------------------------------------
// Kernel C: banded windowed attention per (row h, direction).
// block = 256 (8 waves). Wave handles i-tiles {wave, wave+8} of 16 queries.
// S (16x256) kept in 128 VGPRs; softmax via shfl_xor within 16-lane groups;
// P round-trips through per-wave LDS tile to build A-fragments for P@V WMMA.
// ---------------------------------------------------------------------------
__global__ __launch_bounds__(256)
void scam_attn_kernel(const float* __restrict__ qkv,
                      const float* __restrict__ beta,
                      const float* __restrict__ gamma,
                      float* __restrict__ outbuf) {
    __shared__ float pbuf[8][256];   // 8 KB, one 16x16 tile per wave

    int h   = blockIdx.x;
    int dir = blockIdx.y;            // 0: r2l (Q=q_l,K=k_r,V=v_r,*beta)
                                     // 1: l2r (Q=q_r,K=k_l,V=v_l,*gamma)
    size_t hbase = (size_t)h * Ww * Cc;
    const float* Q  = qkv + (size_t)(dir ? 3 : 0) * CHW + hbase;
    const float* K  = qkv + (size_t)(dir ? 1 : 4) * CHW + hbase;
    const float* V  = qkv + (size_t)(dir ? 2 : 5) * CHW + hbase;
    const float* bg = dir ? gamma : beta;
    float* out = outbuf + (size_t)dir * CHW + (size_t)h * Ww;

    int lane = threadIdx.x & 31;
    int wvid = threadIdx.x >> 5;     // 0..7
    int half = lane >> 4;
    int lm   = lane & 15;
    float* myP = &pbuf[wvid][0];

    for (int iter = 0; iter < 2; ++iter) {
        int it = wvid + iter * 8;    // i-tile 0..15
        int i0 = it * 16;

        // ---- preload Q fragments (A-layout), fixed across j-tiles ----
        v2f qa[16];
        #pragma unroll
        for (int s = 0; s < 16; ++s) {
            int c = 4 * s + 2 * half;
            qa[s].x = Q[(size_t)(i0 + lm) * Cc + c];
            qa[s].y = Q[(size_t)(i0 + lm) * Cc + c + 1];
        }

        // ---- S = Q K^T, 16 j-tiles, K=64 as 16 x (K=4) WMMA steps ----
        v8f S[16];
        #pragma unroll
        for (int jt = 0; jt < 16; ++jt) {
            v8f acc = {};
            #pragma unroll
            for (int s = 0; s < 16; ++s) {
                int c = 4 * s + 2 * half;
                v2f kb;                                  // B(k=c, n=key lm)
                kb.x = K[(size_t)(jt * 16 + lm) * Cc + c];
                kb.y = K[(size_t)(jt * 16 + lm) * Cc + c + 1];
                acc = wmma4(qa[s], kb, acc);
            }
            S[jt] = acc;
        }

        // ---- softmax row stats (rows m = v + 8*half, cols across lanes) ----
        float rmax[8], rsum[8], rinv[8];
        #pragma unroll
        for (int v = 0; v < 8; ++v) rmax[v] = -1e30f;
        #pragma unroll
        for (int jt = 0; jt < 16; ++jt) {
            #pragma unroll
            for (int v = 0; v < 8; ++v) {
                int i = i0 + v + 8 * half;
                int j = jt * 16 + lm;
                bool valid = (i - j <= DWIN) && (j - i <= DWIN);
                float sv = valid ? S[jt][v] * SCALE : -1e30f;
                rmax[v] = fmaxf(rmax[v], sv);
            }
        }
        #pragma unroll
        for (int msk = 1; msk <= 8; msk <<= 1)
            #pragma unroll
            for (int v = 0; v < 8; ++v)
                rmax[v] = fmaxf(rmax[v], __shfl_xor(rmax[v], msk, 32));

        #pragma unroll
        for (int v = 0; v < 8; ++v) rsum[v] = 0.f;
        #pragma unroll
        for (int jt = 0; jt < 16; ++jt) {
            #pragma unroll
            for (int v = 0; v < 8; ++v) {
                int i = i0 + v + 8 * half;
                int j = jt * 16 + lm;
                bool valid = (i - j <= DWIN) && (j - i <= DWIN);
                rsum[v] += valid ? __expf(S[jt][v] * SCALE - rmax[v]) : 0.f;
            }
        }
        #pragma unroll
        for (int msk = 1; msk <= 8; msk <<= 1)
            #pragma unroll
            for (int v = 0; v < 8; ++v)
                rsum[v] += __shfl_xor(rsum[v], msk, 32);
        #pragma unroll
        for (int v = 0; v < 8; ++v) rinv[v] = 1.0f / rsum[v];

        // ---- O = P V, accumulated over j-tiles; P via LDS layout-convert ----
        v8f O[4] = {};
        #pragma unroll
        for (int jt = 0; jt < 16; ++jt) {
            #pragma unroll
            for (int v = 0; v < 8; ++v) {
                int m = v + 8 * half;                    // query row in tile
                int i = i0 + m;
                int j = jt * 16 + lm;
                bool valid = (i - j <= DWIN) && (j - i <= DWIN);
                float p = valid ? __expf(S[jt][v] * SCALE - rmax[v]) * rinv[v] : 0.f;
                myP[m * 16 + lm] = p;                    // C/D layout -> LDS
            }
            // same-wave LDS RAW; compiler inserts s_wait_dscnt
            #pragma unroll
            for (int s4 = 0; s4 < 4; ++s4) {
                v2f a;                                   // A(m=lm, k=4*s4+..)
                a.x = myP[lm * 16 + 4 * s4 + 2 * half + 0];
                a.y = myP[lm * 16 + 4 * s4 + 2 * half + 1];
                int krow = jt * 16 + 4 * s4 + 2 * half;
                #pragma unroll
                for (int t4 = 0; t4 < 4; ++t4) {
                    v2f vb;                              // B(k=key, n=channel)
                    vb.x = V[(size_t)(krow + 0) * Cc + t4 * 16 + lm];
                    vb.y = V[(size_t)(krow + 1) * Cc + t4 * 16 + lm];
                    O[t4] = wmma4(a, vb, O[t4]);
                }
            }
        }

        // ---- store delta = O * (beta|gamma), back to (C,H,W) ----
        #pragma unroll
        for (int t4 = 0; t4 < 4; ++t4) {
            float g = bg[t4 * 16 + lm];
            #pragma unroll
            for (int v = 0; v < 8; ++v) {
                int m = v + 8 * half;                    // query offset
                int c = t4 * 16 + lm;                    // channel
                out[(size_t)c * HWp + i0 + m] = O[t4][v] * g;
            }
        }
    }
}

// ---------------------------------------------------------------------------
extern "C" void kernel_launch(void* const* d_in, const int* in_sizes, int n_in,
                              void* d_out, int out_size, void* d_ws, size_t ws_size,
                              hipStream_t stream) {
    (void)in_sizes; (void)n_in; (void)out_size; (void)ws_size;
    const float* xl    = (const float*)d_in[0];
    const float* xr    = (const float*)d_in[1];
    const float* gl    = (const float*)d_in[2];
    const float* bl    = (const float*)d_in[3];
    const float* gr    = (const float*)d_in[4];
    const float* br    = (const float*)d_in[5];
    const float* Wq    = (const float*)d_in[6];
    const float* bq    = (const float*)d_in[7];
    const float* Wk    = (const float*)d_in[8];
    const float* bk    = (const float*)d_in[9];
    const float* Wv    = (const float*)d_in[10];
    const float* bv    = (const float*)d_in[11];
    const float* beta  = (const float*)d_in[12];
    const float* gamma = (const float*)d_in[13];

    float* ws  = (float*)d_ws;
    float* xn  = ws;                          // 2 * CHW floats
    float* qkv = ws + 2 * (size_t)CHW;        // 6 * CHW floats
    float* out = (float*)d_out;               // delta_l ++ delta_r

    dim3 gA((HWp + 255) / 256, 2);
    scam_ln_kernel<<<gA, dim3(256), 0, stream>>>(xl, xr, gl, bl, gr, br, xn);

    dim3 gB(HWp / 16, 6);
    scam_proj_kernel<<<gB, dim3(128), 0, stream>>>(xl, xr, xn, Wq, bq, Wk, bk, Wv, bv, qkv);

    dim3 gC(Hh, 2);
    scam_attn_kernel<<<gC, dim3(256), 0, stream>>>(qkv, beta, gamma, out);
}